// CSConv_15290083574379
// MI455X (gfx1250) — compile-verified
//
#include <hip/hip_runtime.h>
#include <math.h>

#define BN_EPS 1e-5f

// Problem constants (fixed by the reference setup)
#define BATCH 16
#define CIN   256
#define HW    80
#define COUT  512

#define LDSROWS 82    // 80 rows + 2 halo rows
#define LDSW    88    // LDS row stride in floats (cols shifted +4 for 16B alignment)
#define NTHREADS 320  // 16 row-groups x 20 col-groups

// ---- gfx1250 async global->LDS path (compile-probe guarded) ----------------
#if defined(__HIP_DEVICE_COMPILE__)
#if __has_builtin(__builtin_amdgcn_global_load_async_to_lds_b128)
#define USE_ASYNC_LDS 1
#endif
#endif

typedef int   v4i __attribute__((ext_vector_type(4)));
typedef float v4f __attribute__((ext_vector_type(4)));
typedef __attribute__((address_space(1))) v4i gv4i;   // global int4
typedef __attribute__((address_space(3))) v4i lv4i;   // LDS int4

__device__ __forceinline__ gv4i* as_global_v4(const void* p) {
    // flat global address == addrspace(1) address on amdgcn
    return (gv4i*)(unsigned long long)(uintptr_t)p;
}
__device__ __forceinline__ lv4i* as_local_v4(const void* p) {
    // low 32 bits of a flat LDS address are the LDS byte offset
    return (lv4i*)(unsigned int)(uintptr_t)p;
}

// load 6 floats (cols c0-1 .. c0+4) of LDS row (base + krow*LDSW) into dst[0..5]
// krow is a compile-time constant at every call site -> immediate DS offsets
__device__ __forceinline__ void load_row(const float* lbase, int krow, float* dst) {
    const float* p = lbase + krow * LDSW;
    dst[0] = p[0];                      // ds_load_b32
    const v4f v = *(const v4f*)(p + 1); // ds_load_b128 (16B aligned)
    dst[1] = v.x; dst[2] = v.y; dst[3] = v.z; dst[4] = v.w;
    dst[5] = p[5];                      // ds_load_b32
}

__global__ __launch_bounds__(NTHREADS) void dwconv_bn_silu_kernel(
    const float* __restrict__ x,      // (16,256,80,80)
    const float* __restrict__ w,      // (512,1,3,3)
    const float* __restrict__ bias,   // (512)
    const float* __restrict__ gamma,  // (512)
    const float* __restrict__ beta,   // (512)
    const float* __restrict__ mean,   // (512)
    const float* __restrict__ var,    // (512)
    float* __restrict__ out)          // (16,512,80,80)
{
    __shared__ __align__(16) float tile[LDSROWS * LDSW];   // 28,864 B

    const int tid = threadIdx.x;          // 0..319
    const int ci  = blockIdx.x;           // 0..255 input channel
    const int b   = blockIdx.y;           // 0..15  batch

    const float* xc = x + (size_t)(b * CIN + ci) * (HW * HW);

    // ---- stage whole 82-row haloed channel into LDS -------------------------
    // LDS row lr holds input row gr = lr-1; image col c maps to LDS index c+4
    {
        const int lr0 = tid / 20;          // 0..15  (computed once)
        const int f   = tid % 20;          // float4 chunk within row
        const int fo  = 4 * f;             // float offset of chunk
        for (int lr = lr0; lr < LDSROWS; lr += 16) {
            float* lp = &tile[lr * LDSW + 4 + fo];
            const int gr = lr - 1;         // global input row
            if (gr >= 0 && gr < HW) {
                const float* gp = xc + gr * HW + fo;
#ifdef USE_ASYNC_LDS
                __builtin_amdgcn_global_load_async_to_lds_b128(
                    as_global_v4(gp), as_local_v4(lp), /*offset=*/0, /*cpol=*/0);
#else
                *(v4f*)lp = *(const v4f*)gp;
#endif
            } else {
                *(v4f*)lp = (v4f){0.f, 0.f, 0.f, 0.f};   // top/bottom halo rows
            }
        }
    }
    // left/right halo columns (image col -1 -> idx 3, col 80 -> idx 84): always 0
    if (tid < 2 * LDSROWS) {
        const int lr = tid >> 1;
        tile[lr * LDSW + ((tid & 1) ? 84 : 3)] = 0.0f;
    }

#ifdef USE_ASYNC_LDS
#if __has_builtin(__builtin_amdgcn_s_wait_asynccnt)
    __builtin_amdgcn_s_wait_asynccnt(0);
#else
    asm volatile("s_wait_asynccnt 0" ::: "memory");
#endif
#endif
    __syncthreads();

    // ---- uniform per-channel params (blockIdx-uniform -> scalar loads) ------
    const int co0 = ci * 2;
    float wk[2][9], A[2], C[2];
#pragma unroll
    for (int s = 0; s < 2; ++s) {
        const int co = co0 + s;
#pragma unroll
        for (int k = 0; k < 9; ++k) wk[s][k] = w[co * 9 + k];
        const float a = gamma[co] * rsqrtf(var[co] + BN_EPS);
        A[s] = a;
        C[s] = fmaf(bias[co], a, beta[co] - mean[co] * a);
    }

    // ---- compute: each thread does 5 consecutive rows x 4 cols x 2 channels -
    const int rg = tid / 20;               // 0..15 -> rows rg*5 .. rg*5+4
    const int cg = tid % 20;               // 0..19
    const int c0 = cg * 4;
    const int R0 = rg * 5;

    // single base pointers; all further offsets are compile-time immediates
    const float* lbase = tile + R0 * LDSW + 3 + c0;               // LDS row R0
    float* obase = out + ((size_t)(b * COUT + co0) * HW + R0) * HW + c0;

    // rolling 3-row register window; output row R0+rr uses LDS rows R0+rr .. R0+rr+2
    float rbuf[3][6];
#pragma unroll
    for (int kh = 0; kh < 3; ++kh) load_row(lbase, kh, rbuf[kh]);

#pragma unroll
    for (int rr = 0; rr < 5; ++rr) {
#pragma unroll
        for (int s = 0; s < 2; ++s) {
            float acc[4] = {0.f, 0.f, 0.f, 0.f};
#pragma unroll
            for (int kh = 0; kh < 3; ++kh) {
                const float* rw = rbuf[(rr + kh) % 3];   // row R0+rr+kh (constant idx)
#pragma unroll
                for (int kw = 0; kw < 3; ++kw) {
                    const float wv = wk[s][kh * 3 + kw];
#pragma unroll
                    for (int j = 0; j < 4; ++j)
                        acc[j] = fmaf(wv, rw[j + kw], acc[j]);
                }
            }
            v4f o;
#pragma unroll
            for (int j = 0; j < 4; ++j) {
                const float y = fmaf(acc[j], A[s], C[s]);
                // SiLU: y * sigmoid(y); fast v_rcp_f32 instead of refined divide
                o[j] = y * __builtin_amdgcn_rcpf(1.0f + __expf(-y));
            }
            // written once, never re-read: non-temporal b128 store, immediate offset
            __builtin_nontemporal_store(o, (v4f*)(obase + (size_t)s * (HW * HW) + rr * HW));
        }
        // slide window down: overwrite the oldest row with LDS row R0+rr+3
        if (rr < 4) load_row(lbase, rr + 3, rbuf[rr % 3]);
    }
}

extern "C" void kernel_launch(void* const* d_in, const int* in_sizes, int n_in,
                              void* d_out, int out_size, void* d_ws, size_t ws_size,
                              hipStream_t stream) {
    (void)in_sizes; (void)n_in; (void)out_size; (void)d_ws; (void)ws_size;
    const float* x     = (const float*)d_in[0];
    const float* w     = (const float*)d_in[1];
    const float* bias  = (const float*)d_in[2];
    const float* gamma = (const float*)d_in[3];
    const float* beta  = (const float*)d_in[4];
    const float* mean  = (const float*)d_in[5];
    const float* var   = (const float*)d_in[6];
    float* out = (float*)d_out;

    dim3 grid(CIN, BATCH);   // (256, 16)
    dwconv_bn_silu_kernel<<<grid, NTHREADS, 0, stream>>>(x, w, bias, gamma, beta,
                                                         mean, var, out);
}